// HeadTargetLayer_84413287235742
// MI455X (gfx1250) — compile-verified
//
#include <hip/hip_runtime.h>

#define N_IMG 32
#define L_ROI 4096
#define M_GT  128
#define C_CLS 80
#define IOU_TH 0.5f
#define BLK 256

// ---------------------------------------------------------------------------
// Kernel 1: IoU match + per-roi NLL.
// One block handles 256 rois of one image. The image's 128 gt boxes (2 KB)
// are staged into LDS with CDNA5 async-LDS DMA (ASYNCcnt / s_wait_asynccnt),
// then reused by all 256 rois of this block.
// ---------------------------------------------------------------------------
__global__ __launch_bounds__(BLK) void k_match_nll(
    const float* __restrict__ rois,        // [N,L,4]
    const float* __restrict__ cls_scores,  // [N,L,C]
    const float* __restrict__ gt_boxes,    // [N,M,4]
    const int*   __restrict__ gt_clses,    // [N,M]
    float* __restrict__ nll_out,           // [N,L]
    int*   __restrict__ lab_out,           // [N,L]
    int*   __restrict__ pos_out)           // [N,L]
{
    __shared__ float4 s_gt[M_GT];
    __shared__ int    s_cls[M_GT];

    const int chunks = L_ROI / BLK;          // 16
    const int n      = blockIdx.x / chunks;
    const int chunk  = blockIdx.x % chunks;
    const int tid    = threadIdx.x;
    const int l      = chunk * BLK + tid;

    // --- CDNA5 async DMA: global -> LDS, one b128 per lane (waves 0..3) ---
    if (tid < M_GT) {
        const float* gsrc = gt_boxes + ((size_t)n * M_GT + tid) * 4;
        unsigned ldsoff = (unsigned)(size_t)(&s_gt[tid]);   // low 32b of flat LDS addr = LDS offset
        asm volatile("global_load_async_to_lds_b128 %0, %1, off"
                     :: "v"(ldsoff), "v"(gsrc) : "memory");
    }
    if (tid >= 128 && tid < 128 + M_GT) {
        s_cls[tid - 128] = gt_clses[(size_t)n * M_GT + (tid - 128)];
    }
    asm volatile("s_wait_asynccnt 0" ::: "memory");
    __syncthreads();

    // Prefetch my score row (global_prefetch_b8) so it lands while we do IoU.
    const float* srow = cls_scores + ((size_t)n * L_ROI + l) * C_CLS;
    __builtin_prefetch(srow, 0, 1);

    // --- IoU argmax over M=128 gts (LDS-resident, broadcast reads) ---
    const float4 rb = *(const float4*)(rois + ((size_t)n * L_ROI + l) * 4);
    const float area_r = (rb.z - rb.x) * (rb.w - rb.y);

    float best_iou = -1.0f;
    int   best_m   = 0;
#pragma unroll 4
    for (int m = 0; m < M_GT; ++m) {
        const float4 g = s_gt[m];
        const float lx = fmaxf(rb.x, g.x), ly = fmaxf(rb.y, g.y);
        const float rx = fminf(rb.z, g.z), ry = fminf(rb.w, g.w);
        const float w  = fmaxf(rx - lx, 0.0f), h = fmaxf(ry - ly, 0.0f);
        const float inter  = w * h;
        const float area_g = (g.z - g.x) * (g.w - g.y);
        const float iou    = inter / (area_r + area_g - inter);
        if (iou > best_iou) { best_iou = iou; best_m = m; }   // first-max, like jnp.argmax
    }
    const int label = s_cls[best_m];

    // --- log-softmax NLL over C=80 (register-resident row, float4 loads) ---
    const float4* srow4 = (const float4*)srow;
    float4 v[C_CLS / 4];
#pragma unroll
    for (int i = 0; i < C_CLS / 4; ++i) v[i] = srow4[i];
    float mx = v[0].x;
#pragma unroll
    for (int i = 0; i < C_CLS / 4; ++i) {
        mx = fmaxf(mx, fmaxf(fmaxf(v[i].x, v[i].y), fmaxf(v[i].z, v[i].w)));
    }
    float s = 0.0f;
#pragma unroll
    for (int i = 0; i < C_CLS / 4; ++i) {
        s += expf(v[i].x - mx) + expf(v[i].y - mx) + expf(v[i].z - mx) + expf(v[i].w - mx);
    }
    const float lse = mx + logf(s);
    const float nll = lse - srow[label];   // dynamic index -> single global load, no spill

    const size_t o = (size_t)n * L_ROI + l;
    nll_out[o] = nll;
    lab_out[o] = label;
    pos_out[o] = (best_iou >= IOU_TH) ? 1 : 0;
}

// ---------------------------------------------------------------------------
// Kernel 2: per-image balanced sampling + mean CE. One block per image.
// Stage the image's 4096-entry nll/label/pos into LDS (48 KB of 320 KB WGP
// pool), count positives in parallel, then an in-order scan with per-class
// counters reproduces the exact "first ceil(max(1,n_pos/C)) per class in
// roi-index order" semantics.
// ---------------------------------------------------------------------------
__global__ __launch_bounds__(BLK) void k_select_loss(
    const float* __restrict__ nll_in,
    const int*   __restrict__ lab_in,
    const int*   __restrict__ pos_in,
    float* __restrict__ per_img)           // [N]
{
    __shared__ float s_nll[L_ROI];
    __shared__ int   s_lab[L_ROI];
    __shared__ int   s_pos[L_ROI];
    __shared__ int   s_cnt[C_CLS];
    __shared__ int   s_npos;

    const int n = blockIdx.x, tid = threadIdx.x;
    if (tid == 0) s_npos = 0;
    if (tid < C_CLS) s_cnt[tid] = 0;
    __syncthreads();

    int local = 0;
    for (int l = tid; l < L_ROI; l += BLK) {
        const size_t o = (size_t)n * L_ROI + l;
        const int p = pos_in[o];
        s_nll[l] = nll_in[o];
        s_lab[l] = lab_in[o];
        s_pos[l] = p;
        local += p;
    }
    atomicAdd(&s_npos, local);
    __syncthreads();

    if (tid == 0) {
        const int np  = s_npos;
        const int smp = (int)ceilf(fmaxf(1.0f, (float)np / (float)C_CLS));
        float sum = 0.0f;
        int   cnt = 0;
        for (int l = 0; l < L_ROI; ++l) {
            if (s_pos[l]) {
                const int c = s_lab[l];
                if (s_cnt[c]++ < smp) { sum += s_nll[l]; ++cnt; }
            }
        }
        per_img[n] = sum / (float)cnt;   // 0/0 -> NaN matches reference when no positives
    }
}

// ---------------------------------------------------------------------------
// Kernel 3: sum 32 per-image means -> scalar loss.
// ---------------------------------------------------------------------------
__global__ void k_finalize(const float* __restrict__ per_img, float* __restrict__ out)
{
    if (threadIdx.x == 0 && blockIdx.x == 0) {
        float s = 0.0f;
        for (int n = 0; n < N_IMG; ++n) s += per_img[n];
        out[0] = s;
    }
}

// ---------------------------------------------------------------------------
extern "C" void kernel_launch(void* const* d_in, const int* in_sizes, int n_in,
                              void* d_out, int out_size, void* d_ws, size_t ws_size,
                              hipStream_t stream)
{
    (void)in_sizes; (void)n_in; (void)out_size; (void)ws_size;
    const float* rois = (const float*)d_in[0];   // [N,L,4]
    const float* cls  = (const float*)d_in[1];   // [N,L,C]
    // d_in[2] bbox_deltas: multiplied by 0.0 in the reference -> value-neutral, skipped
    const float* gt   = (const float*)d_in[3];   // [N,M,4]
    const int*   gcls = (const int*)d_in[4];     // [N,M]

    char* ws = (char*)d_ws;
    const size_t nl = (size_t)N_IMG * L_ROI;
    float* nll = (float*)(ws);
    int*   lab = (int*)(ws + nl * 4);
    int*   pos = (int*)(ws + 2 * nl * 4);
    float* per = (float*)(ws + 3 * nl * 4);

    k_match_nll<<<N_IMG * (L_ROI / BLK), BLK, 0, stream>>>(rois, cls, gt, gcls, nll, lab, pos);
    k_select_loss<<<N_IMG, BLK, 0, stream>>>(nll, lab, pos, per);
    k_finalize<<<1, 32, 0, stream>>>(per, (float*)d_out);
}